// KnnNCELoss_78666620993994
// MI455X (gfx1250) — compile-verified
//
#include <hip/hip_runtime.h>
#include <hip/hip_bf16.h>
#include <stdint.h>

#define N_S   1024
#define D_F   256
#define KNN   16
#define DSTR  1028           // padded LDS row stride (floats) for distance block
#define TJ    64             // j-tile width

typedef float v2f __attribute__((ext_vector_type(2)));
typedef float v8f __attribute__((ext_vector_type(8)));

// ---- CDNA5 async global->LDS copy (inline asm: portable across toolchains) ----
__device__ __forceinline__ void async_ld_b128(const void* g, void* l) {
    // generic pointer to a __shared__ object: low 32 bits == LDS offset (aperture in high bits)
    unsigned       loff = (unsigned)(unsigned long long)(uintptr_t)l;
    unsigned long long ga = (unsigned long long)(uintptr_t)g;
    asm volatile("global_load_async_to_lds_b128 %0, %1, off"
                 :: "v"(loff), "v"(ga) : "memory");
}
__device__ __forceinline__ void wait_async0() {
    asm volatile("s_wait_asynccnt 0x0" ::: "memory");
}

// ---- kernel 1: yT[f][i] = w[f] * x[i][f]  (w>=0 so w*|a-b| == |wa-wb|) ----
__global__ __launch_bounds__(256) void prep_kernel(const float* __restrict__ x,
                                                   const float* __restrict__ w,
                                                   float* __restrict__ yT) {
    int id = blockIdx.x * 256 + threadIdx.x;      // 256K elements
    int f = id >> 10, i = id & (N_S - 1);
    yT[id] = w[f] * x[i * D_F + f];
}

// ---- kernel 2: per 16-row block: distances, WMMA label-GEMM, top-16, row loss ----
__global__ __launch_bounds__(256) void wllt_main(const float* __restrict__ yT,
                                                 const int*   __restrict__ lab,
                                                 float* __restrict__ row_loss) {
    __shared__ float tileS[D_F * TJ];     // 64 KB  staged yT tile [f][64]
    __shared__ float Dl[16 * DSTR];       // 65.8KB distance block, padded rows
    __shared__ float yrowT[D_F * 16];     // 16 KB  this block's rows, [f][r]
    __shared__ int   labS[N_S];           // 4 KB
    __shared__ float Cws[8 * 256];        // 8 KB   per-wave WMMA C partials
    __shared__ float Csum[256];           // 1 KB

    const int tid  = threadIdx.x;
    const int lane = tid & 31;
    const int wv   = tid >> 5;            // 8 waves
    const int i0   = blockIdx.x * 16;

    for (int e = tid; e < N_S; e += 256) labS[e] = lab[e];
    for (int e = tid; e < D_F * 16; e += 256) {
        int f = e >> 4, r = e & 15;
        yrowT[e] = yT[f * N_S + i0 + r];  // already w-scaled
    }
    __syncthreads();

    // ---------- phase 1: D[r][j] = sum_f |y_r[f] - y_j[f]| ----------
    for (int t = 0; t < 16; ++t) {        // 16 tiles of 64 columns
        // async-stage tile: 4096 float4, 16 per thread
        #pragma unroll
        for (int r = 0; r < 16; ++r) {
            int e = r * 256 + tid;        // float4 slot
            int f = e >> 4, c4 = e & 15;
            async_ld_b128(yT + f * N_S + t * TJ + c4 * 4, &tileS[e * 4]);
        }
        wait_async0();
        __syncthreads();

        // wave wv owns rows 2wv, 2wv+1; lane owns cols 2*lane, 2*lane+1
        float a00 = 0.f, a01 = 0.f, a10 = 0.f, a11 = 0.f;
        #pragma unroll 4
        for (int f = 0; f < D_F; ++f) {
            float2 yr = *(const float2*)&yrowT[f * 16 + 2 * wv];   // broadcast
            float2 yj = *(const float2*)&tileS[f * TJ + 2 * lane]; // conflict-free b64
            a00 += fabsf(yr.x - yj.x); a01 += fabsf(yr.x - yj.y);
            a10 += fabsf(yr.y - yj.x); a11 += fabsf(yr.y - yj.y);
        }
        int jb = t * TJ + 2 * lane;
        *(float2*)&Dl[(2 * wv)     * DSTR + jb] = make_float2(a00, a01);
        *(float2*)&Dl[(2 * wv + 1) * DSTR + jb] = make_float2(a10, a11);
        __syncthreads();                  // tile reused next iteration
    }

    // ---------- phase 2: C = K_block @ [onehot(lab) | ones | 0]  via WMMA ----------
    // k_sum_diff_i = C[i][10] - C[i][lab_i]   (diagonal cancels exactly)
    {
        const int m   = lane & 15;        // A: row M / B: col N
        const int sel = lane >> 4;        // K-halfwave select (+2)
        v8f c = {};
        for (int ch = wv; ch < 256; ch += 8) {   // K-dim chunks of 4 (uniform per wave)
            int col = (ch << 2) + (sel << 1);
            float2 dp = *(const float2*)&Dl[m * DSTR + col];   // conflict-free b64
            v2f a; a.x = __expf(-dp.x); a.y = __expf(-dp.y);   // A = K tile
            int2 lp = *(const int2*)&labS[col];
            v2f b;
            b.x = (m == lp.x || m == 10) ? 1.f : 0.f;          // B = [H | 1 | 0]
            b.y = (m == lp.y || m == 10) ? 1.f : 0.f;
            c = __builtin_amdgcn_wmma_f32_16x16x4_f32(false, a, false, b,
                                                      (short)0, c, false, false);
        }
        #pragma unroll
        for (int v = 0; v < 8; ++v) Cws[wv * 256 + v * 32 + lane] = c[v];
    }
    __syncthreads();
    {   // reduce 8 wave-partials, fixed order (deterministic)
        float s = 0.f;
        #pragma unroll
        for (int ww = 0; ww < 8; ++ww) s += Cws[ww * 256 + tid];
        Csum[tid] = s;
    }
    __syncthreads();

    // ---------- phase 3: per-row top-16 smallest same-label distances ----------
    for (int rr = wv; rr < 16; rr += 8) {
        const int   ig   = i0 + rr;
        const int   labr = labS[ig];
        float tk[KNN];
        #pragma unroll
        for (int q = 0; q < KNN; ++q) tk[q] = 1e30f;
        float tmax = 1e30f;

        for (int j = lane; j < N_S; j += 32) {
            if (labS[j] == labr && j != ig) {
                float d = Dl[rr * DSTR + j];
                if (d < tmax) {
                    int am = 0; float mv = tk[0];
                    #pragma unroll
                    for (int q = 1; q < KNN; ++q) if (tk[q] > mv) { mv = tk[q]; am = q; }
                    #pragma unroll
                    for (int q = 0; q < KNN; ++q) if (q == am) tk[q] = d;
                    mv = tk[0];
                    #pragma unroll
                    for (int q = 1; q < KNN; ++q) mv = fmaxf(mv, tk[q]);
                    tmax = mv;
                }
            }
        }
        // merge: 16 rounds of wave32 min-extract
        float sum_d = 0.f, ksame = 0.f; int cnt = 0;
        for (int s = 0; s < KNN; ++s) {
            float m = tk[0];
            #pragma unroll
            for (int q = 1; q < KNN; ++q) m = fminf(m, tk[q]);
            float g = m;
            #pragma unroll
            for (int off = 16; off > 0; off >>= 1) g = fminf(g, __shfl_xor(g, off, 32));
            if (g < 1e29f) {
                sum_d += g; ksame += __expf(-g); ++cnt;
                unsigned long long bal = __ballot(m == g);
                int first = (int)__ffsll((long long)bal) - 1;
                if (lane == first) {
                    bool done = false;
                    #pragma unroll
                    for (int q = 0; q < KNN; ++q)
                        if (!done && tk[q] == m) { tk[q] = 1e30f; done = true; }
                }
            }
        }
        if (lane == 0) {
            int base = (rr & 7) * 32 + ((rr >> 3) << 4);
            float ksd = Csum[base + 10] - Csum[base + labr];   // k_sum_diff_i
            float lt  = logf(fmaxf(ksame + ksd, 1e-10f));
            row_loss[ig] = sum_d + (float)cnt * lt;
        }
    }
}

// ---- kernel 3: deterministic final reduction ----
__global__ __launch_bounds__(256) void reduce_kernel(const float* __restrict__ rl,
                                                     float* __restrict__ out) {
    __shared__ float s[256];
    float a = 0.f;
    for (int k = threadIdx.x; k < N_S; k += 256) a += rl[k];
    s[threadIdx.x] = a;
    __syncthreads();
    for (int st = 128; st > 0; st >>= 1) {
        if (threadIdx.x < st) s[threadIdx.x] += s[threadIdx.x + st];
        __syncthreads();
    }
    if (threadIdx.x == 0) out[0] = s[0] / ((float)N_S * (float)N_S);
}

extern "C" void kernel_launch(void* const* d_in, const int* in_sizes, int n_in,
                              void* d_out, int out_size, void* d_ws, size_t ws_size,
                              hipStream_t stream) {
    const float* samples = (const float*)d_in[0];   // [1024,256] f32
    const float* weights = (const float*)d_in[1];   // [256] f32
    const int*   labels  = (const int*)d_in[2];     // [1024] i32
    float* out = (float*)d_out;

    float* yT       = (float*)d_ws;                 // 256*1024 f32 = 1 MB
    float* row_loss = yT + D_F * N_S;               // 1024 f32

    prep_kernel<<<(D_F * N_S) / 256, 256, 0, stream>>>(samples, weights, yT);
    wllt_main<<<N_S / 16, 256, 0, stream>>>(yT, labels, row_loss);
    reduce_kernel<<<1, 256, 0, stream>>>(row_loss, out);
}